// Model_55319178772709
// MI455X (gfx1250) — compile-verified
//
#include <hip/hip_runtime.h>

// ---------------- problem constants ----------------
#define B_    32
#define N_    131072
#define CPB   32                 // chunks per batch row
#define SEG   (N_/CPB)           // 4096 points per block
#define NBLK  (B_*CPB)           // 1024 blocks per streaming pass
#define TPB   256
#define EPS_  1e-5f

// ---------------- workspace layout (floats) ----------------
#define WS_PART  0                    // NBLK*6 partial (sum[3],sumsq[3])
#define WS_MAXP  (NBLK*6)             // NBLK partial maxes
#define WS_STATS (WS_MAXP + NBLK)     // 6 BN layers * (scale[3],shift[3]) = 36
#define WS_GF0   (WS_STATS + 36)      // 32
#define WS_ROT   (WS_GF0 + 32)        // 9

#define Y_SZ   ((size_t)B_*(size_t)N_)   // 4194304
#define YC_OFF (Y_SZ)
#define XC_OFF (Y_SZ*4)

// ---------------- CDNA5 feature guards ----------------
#if __has_builtin(__builtin_amdgcn_global_load_async_to_lds_b128) && __has_builtin(__builtin_amdgcn_s_wait_asynccnt)
#define HAVE_ASYNC 1
#endif
#if __has_builtin(__builtin_amdgcn_wmma_f32_16x16x4_f32)
#define HAVE_WMMA4 1
#endif

typedef float v2f __attribute__((ext_vector_type(2)));
typedef float v8f __attribute__((ext_vector_type(8)));
typedef int   v4i __attribute__((ext_vector_type(4)));
typedef __attribute__((address_space(1))) v4i* gv4i_p;   // global 16B vector ptr
typedef __attribute__((address_space(3))) v4i* lv4i_p;   // LDS    16B vector ptr

struct Consts {
  const float* x;
  const float *c_w0,*c_b0,*c_w1,*c_b1,*c_w2,*c_b2;       // cb0
  const float *s1_w0,*s1_b0,*s1_w1,*s1_b1,*s1_w2,*s1_b2; // sb1
  const float *s2_w0,*s2_b0,*s2_w1,*s2_b1,*s2_w2,*s2_b2; // sb2
  const float *cls_w,*cls_b;
  float* ws;
  float* out;
};

// ---------------- small device helpers ----------------
__device__ __forceinline__ float relu_(float v){ return v > 0.f ? v : 0.f; }
__device__ __forceinline__ void ld_(float* d, const float* s, int n){
  #pragma unroll
  for(int i=0;i<n;++i) d[i]=s[i];
}
__device__ __forceinline__ void mv33_(const float* w, const float* b, const float* v, float* r){
  r[0]=fmaf(w[0],v[0],fmaf(w[1],v[1],fmaf(w[2],v[2],b[0])));
  r[1]=fmaf(w[3],v[0],fmaf(w[4],v[1],fmaf(w[5],v[2],b[1])));
  r[2]=fmaf(w[6],v[0],fmaf(w[7],v[1],fmaf(w[8],v[2],b[2])));
}
__device__ __forceinline__ void bnrelu_(float* v, const float* st){ // st = scale[3],shift[3]
  v[0]=relu_(fmaf(v[0],st[0],st[3]));
  v[1]=relu_(fmaf(v[1],st[1],st[4]));
  v[2]=relu_(fmaf(v[2],st[2],st[5]));
}

// =====================================================================
// Streaming stats / max passes.
// STAGE 0: stats of cb0.conv0(x)           STAGE 1: stats of cb0.conv1(h)
// STAGE 2: max of lf0 (gf0 partials)       STAGE 3: stats of sb1.conv0(x1)
// STAGE 4: stats of sb1.conv1(h)           STAGE 5: stats of sb2.conv0(x2)
// STAGE 6: stats of sb2.conv1(h)
// =====================================================================
template<int STAGE>
__global__ __launch_bounds__(TPB) void k_pass(Consts K)
{
  __shared__ float red[TPB];
  const int tid = threadIdx.x;
  const int b   = blockIdx.x / CPB;
  const int ck  = blockIdx.x % CPB;
  const size_t base = (size_t)b*2u*N_ + (size_t)ck*SEG;
  const float4* __restrict__ q0 = (const float4*)(K.x + base);        // channel 0
  const float4* __restrict__ q1 = (const float4*)(K.x + base + N_);   // channel 1
  const float* ST = K.ws + WS_STATS;

  float cw0[6],cb0v[3],st0[6],cw1[9],cb1v[3],st1[6],cw2[3],cb2v=0.f;
  float gf=0.f, aw0[9],ab0[3],st2[6],aw1[9],ab1[3],st3[6],aw2[9],ab2[3];
  float bw0[9],bb0[3],st4[6],bw1[9],bb1[3];

  if constexpr (STAGE<=2){ ld_(cw0,K.c_w0,6); ld_(cb0v,K.c_b0,3); }
  if constexpr (STAGE>=1 && STAGE<=2){ ld_(st0,ST+0,6); ld_(cw1,K.c_w1,9); ld_(cb1v,K.c_b1,3); }
  if constexpr (STAGE==2){ ld_(st1,ST+6,6); ld_(cw2,K.c_w2,3); cb2v=K.c_b2[0]; }
  if constexpr (STAGE>=3){ gf=K.ws[WS_GF0+b]; ld_(aw0,K.s1_w0,9); ld_(ab0,K.s1_b0,3); }
  if constexpr (STAGE>=4){ ld_(st2,ST+12,6); ld_(aw1,K.s1_w1,9); ld_(ab1,K.s1_b1,3); }
  if constexpr (STAGE>=5){ ld_(st3,ST+18,6); ld_(aw2,K.s1_w2,9); ld_(ab2,K.s1_b2,3);
                           ld_(bw0,K.s2_w0,9); ld_(bb0,K.s2_b0,3); }
  if constexpr (STAGE>=6){ ld_(st4,ST+24,6); ld_(bw1,K.s2_w1,9); ld_(bb1,K.s2_b1,3); }

  float acc[6]={0.f,0.f,0.f,0.f,0.f,0.f};
  float mx = -3.0e38f;

  auto accum=[&](const float* t){
    acc[0]+=t[0]; acc[1]+=t[1]; acc[2]+=t[2];
    acc[3]=fmaf(t[0],t[0],acc[3]);
    acc[4]=fmaf(t[1],t[1],acc[4]);
    acc[5]=fmaf(t[2],t[2],acc[5]);
  };
  auto body=[&](float x0, float x1c){
    if constexpr (STAGE<=2){
      float t[3];
      t[0]=fmaf(cw0[0],x0,fmaf(cw0[1],x1c,cb0v[0]));
      t[1]=fmaf(cw0[2],x0,fmaf(cw0[3],x1c,cb0v[1]));
      t[2]=fmaf(cw0[4],x0,fmaf(cw0[5],x1c,cb0v[2]));
      if constexpr (STAGE==0){ accum(t); return; }
      bnrelu_(t, st0);
      float u[3]; mv33_(cw1,cb1v,t,u);
      if constexpr (STAGE==1){ accum(u); return; }
      bnrelu_(u, st1);
      mx = fmaxf(mx, fmaf(cw2[0],u[0],fmaf(cw2[1],u[1],fmaf(cw2[2],u[2],cb2v))));
    } else {
      float v[3]={x0,x1c,gf};
      float t[3]; mv33_(aw0,ab0,v,t);
      if constexpr (STAGE==3){ accum(t); return; }
      bnrelu_(t, st2);
      float u[3]; mv33_(aw1,ab1,t,u);
      if constexpr (STAGE==4){ accum(u); return; }
      bnrelu_(u, st3);
      float lf1[3]; mv33_(aw2,ab2,u,lf1);
      float x2[3]={v[0]+lf1[0], v[1]+lf1[1], v[2]+lf1[2]};
      float p1[3]; mv33_(bw0,bb0,x2,p1);
      if constexpr (STAGE==5){ accum(p1); return; }
      bnrelu_(p1, st4);
      float p2[3]; mv33_(bw1,bb1,p1,p2);
      accum(p2);
    }
  };

  constexpr int ITER = SEG/(TPB*4);  // 4
  #pragma unroll
  for(int it=0; it<ITER; ++it){
    float4 a = q0[it*TPB+tid];
    float4 c = q1[it*TPB+tid];
    __builtin_prefetch(q0+(it+1)*TPB+tid, 0, 3);   // global_prefetch_b8 (speculative)
    __builtin_prefetch(q1+(it+1)*TPB+tid, 0, 3);
    body(a.x,c.x); body(a.y,c.y); body(a.z,c.z); body(a.w,c.w);
  }

  if constexpr (STAGE==2){
    red[tid]=mx; __syncthreads();
    for(int s=TPB/2;s>0;s>>=1){ if(tid<s) red[tid]=fmaxf(red[tid],red[tid+s]); __syncthreads(); }
    if(tid==0) K.ws[WS_MAXP + blockIdx.x] = red[0];
  } else {
    #pragma unroll
    for(int q=0;q<6;++q){
      red[tid]=acc[q]; __syncthreads();
      for(int s=TPB/2;s>0;s>>=1){ if(tid<s) red[tid]+=red[tid+s]; __syncthreads(); }
      if(tid==0) K.ws[WS_PART + blockIdx.x*6 + q] = red[0];
      __syncthreads();
    }
  }
}

// ----- deterministic finalize: partials -> BN scale/shift -----
__global__ __launch_bounds__(TPB) void k_fin_bn(float* ws, const float* g, const float* be, int layer)
{
  __shared__ float red[TPB];
  __shared__ float tot[6];
  const int tid=threadIdx.x;
  float s[6]={0.f,0.f,0.f,0.f,0.f,0.f};
  for(int i=tid;i<NBLK;i+=TPB){
    const float* p = ws + WS_PART + i*6;
    #pragma unroll
    for(int q=0;q<6;++q) s[q]+=p[q];
  }
  for(int q=0;q<6;++q){
    red[tid]=s[q]; __syncthreads();
    for(int st=TPB/2;st>0;st>>=1){ if(tid<st) red[tid]+=red[tid+st]; __syncthreads(); }
    if(tid==0) tot[q]=red[0];
    __syncthreads();
  }
  if(tid<3){
    const float inv = 1.0f/(float)((long long)B_*N_);   // exact (power of two)
    float mean = tot[tid]*inv;
    float var  = fmaf(-mean, mean, tot[3+tid]*inv);
    var = fmaxf(var, 0.f);
    float sc = g[tid]*rsqrtf(var + EPS_);
    ws[WS_STATS + layer*6 + tid]     = sc;
    ws[WS_STATS + layer*6 + 3 + tid] = fmaf(-mean, sc, be[tid]);
  }
}

__global__ void k_fin_gf0(float* ws)
{
  const int b = threadIdx.x;
  if(b < B_){
    float m=-3.0e38f;
    for(int k=0;k<CPB;++k) m=fmaxf(m, ws[WS_MAXP + b*CPB + k]);
    ws[WS_GF0+b]=m;
  }
}

// =====================================================================
// rot = pinv(onehot(origin)) @ onehot(target) : row-normalized counts.
// Craw = enc_o^T @ enc_t computed with V_WMMA_F32_16X16X4_F32 (K=32 as
// eight chained K=4 WMMAs).  A[m,k]=1[origin[k]==m], B[k,n]=1[target[k]==n].
// One wave, EXEC all-ones, no divergence before the WMMAs.
// =====================================================================
__global__ void k_rot(const int* __restrict__ origin, const int* __restrict__ target, float* ws)
{
  const int lane = threadIdx.x;        // launched with 32 threads
  const int m    = lane & 15;
  const int kb   = (lane < 16) ? 0 : 2;
  float cr0, cr1, cr2;
#ifdef HAVE_WMMA4
  v8f c = {0.f,0.f,0.f,0.f,0.f,0.f,0.f,0.f};
  #pragma unroll
  for(int kk=0; kk<8; ++kk){
    const int j0 = kk*4 + kb;
    v2f a, bbv;
    a[0]   = (origin[j0]   == m) ? 1.f : 0.f;   // A 16x4: lanes<16 K=0,1; lanes>=16 K=2,3
    a[1]   = (origin[j0+1] == m) ? 1.f : 0.f;
    bbv[0] = (target[j0]   == m) ? 1.f : 0.f;   // B 4x16: same K split, n = lane%16
    bbv[1] = (target[j0+1] == m) ? 1.f : 0.f;
    c = __builtin_amdgcn_wmma_f32_16x16x4_f32(false, a, false, bbv, (short)0, c, false, false);
  }
  cr0 = c[0]; cr1 = c[1]; cr2 = c[2];   // D rows 0..2 live in vgprs 0..2 on lanes 0..15
#else
  cr0=cr1=cr2=0.f;
  for(int j=0;j<B_;++j){
    float td = (target[j]==m)?1.f:0.f;
    if(origin[j]==0) cr0+=td;
    if(origin[j]==1) cr1+=td;
    if(origin[j]==2) cr2+=td;
  }
#endif
  float cnt0=__shfl(cr0,0)+__shfl(cr0,1)+__shfl(cr0,2);
  float cnt1=__shfl(cr1,0)+__shfl(cr1,1)+__shfl(cr1,2);
  float cnt2=__shfl(cr2,0)+__shfl(cr2,1)+__shfl(cr2,2);
  if(lane<3){
    ws[WS_ROT+0*3+lane] = cnt0>0.f ? cr0/cnt0 : 0.f;   // pinv limit: zero row if class absent
    ws[WS_ROT+1*3+lane] = cnt1>0.f ? cr1/cnt1 : 0.f;
    ws[WS_ROT+2*3+lane] = cnt2>0.f ? cr2/cnt2 : 0.f;
  }
}

// =====================================================================
// Final fused pass: x -> x1 -> lf1 -> x2 -> lf2 -> x3 -> y3 -> outputs.
// x streamed through LDS via async global->LDS copies (each wave loads
// and consumes only its own slice -> only s_wait_asynccnt needed).
// =====================================================================
__global__ __launch_bounds__(TPB) void k_final(Consts K)
{
  __shared__ float lx[2*SEG];                     // 32 KB: [0..SEG)=ch0, [SEG..2SEG)=ch1
  const int tid=threadIdx.x, lane=tid&31, wave=tid>>5;
  const int b = blockIdx.x / CPB, ck = blockIdx.x % CPB;
  const size_t base = (size_t)b*2u*N_ + (size_t)ck*SEG;
  const float* __restrict__ gx = K.x + base;
  const int wp = wave*(SEG/8);                    // 512 points per wave

#ifdef HAVE_ASYNC
  #pragma unroll
  for(int i=0;i<(SEG/8)/(32*4);++i){              // 4 iters, b128 per lane
    int p = wp + (i*32+lane)*4;
    __builtin_amdgcn_global_load_async_to_lds_b128(
        (gv4i_p)(gx + p),      (lv4i_p)(&lx[p]),     0, 0);
    __builtin_amdgcn_global_load_async_to_lds_b128(
        (gv4i_p)(gx + N_ + p), (lv4i_p)(&lx[SEG+p]), 0, 0);
  }
  __builtin_amdgcn_s_wait_asynccnt(0);
#else
  #pragma unroll
  for(int i=0;i<(SEG/8)/(32*4);++i){
    int p = wp + (i*32+lane)*4;
    *(float4*)&lx[p]     = *(const float4*)(gx + p);
    *(float4*)&lx[SEG+p] = *(const float4*)(gx + N_ + p);
  }
#endif

  const float* ST = K.ws + WS_STATS;
  const float gf = K.ws[WS_GF0+b];
  float aw0[9],ab0[3],st2[6],aw1[9],ab1[3],st3[6],aw2[9],ab2[3];
  float bw0[9],bb0[3],st4[6],bw1[9],bb1[3],st5[6],bw2[9],bb2[3];
  float rot[9],clw[9],clb[3];
  ld_(aw0,K.s1_w0,9); ld_(ab0,K.s1_b0,3); ld_(st2,ST+12,6);
  ld_(aw1,K.s1_w1,9); ld_(ab1,K.s1_b1,3); ld_(st3,ST+18,6);
  ld_(aw2,K.s1_w2,9); ld_(ab2,K.s1_b2,3);
  ld_(bw0,K.s2_w0,9); ld_(bb0,K.s2_b0,3); ld_(st4,ST+24,6);
  ld_(bw1,K.s2_w1,9); ld_(bb1,K.s2_b1,3); ld_(st5,ST+30,6);
  ld_(bw2,K.s2_w2,9); ld_(bb2,K.s2_b2,3);
  ld_(rot,K.ws+WS_ROT,9); ld_(clw,K.cls_w,9); ld_(clb,K.cls_b,3);

  auto pt=[&](float x0, float x1c, float& y, float* yc, float* xc){
    float v[3]={x0,x1c,gf};
    float t[3];  mv33_(aw0,ab0,v,t);  bnrelu_(t,st2);
    float u[3];  mv33_(aw1,ab1,t,u);  bnrelu_(u,st3);
    float lf1[3]; mv33_(aw2,ab2,u,lf1);
    float x2[3]={v[0]+lf1[0], v[1]+lf1[1], v[2]+lf1[2]};
    float p1[3]; mv33_(bw0,bb0,x2,p1); bnrelu_(p1,st4);
    float p2[3]; mv33_(bw1,bb1,p1,p2); bnrelu_(p2,st5);
    float lf2[3]; mv33_(bw2,bb2,p2,lf2);
    float x3[3]={x2[0]+lf2[0], x2[1]+lf2[1], x2[2]+lf2[2]};
    float y3[3];
    y3[0]=fmaf(x3[0],rot[0],fmaf(x3[1],rot[3],x3[2]*rot[6]));
    y3[1]=fmaf(x3[0],rot[1],fmaf(x3[1],rot[4],x3[2]*rot[7]));
    y3[2]=fmaf(x3[0],rot[2],fmaf(x3[1],rot[5],x3[2]*rot[8]));
    y = y3[2]-lf2[2]-lf1[2];                         // y1 channel IN_DIM..(=2)
    yc[0]=fmaf(clw[0],y3[0],fmaf(clw[1],y3[1],fmaf(clw[2],y3[2],clb[0])));
    yc[1]=fmaf(clw[3],y3[0],fmaf(clw[4],y3[1],fmaf(clw[5],y3[2],clb[1])));
    yc[2]=fmaf(clw[6],y3[0],fmaf(clw[7],y3[1],fmaf(clw[8],y3[2],clb[2])));
    xc[0]=fmaf(clw[0],x3[0],fmaf(clw[1],x3[1],fmaf(clw[2],x3[2],clb[0])));
    xc[1]=fmaf(clw[3],x3[0],fmaf(clw[4],x3[1],fmaf(clw[5],x3[2],clb[1])));
    xc[2]=fmaf(clw[6],x3[0],fmaf(clw[7],x3[1],fmaf(clw[8],x3[2],clb[2])));
  };

  float* oy  = K.out +            (size_t)b*N_      + (size_t)ck*SEG;
  float* oyc = K.out + YC_OFF + (size_t)b*3u*N_     + (size_t)ck*SEG;
  float* oxc = K.out + XC_OFF + (size_t)b*3u*N_     + (size_t)ck*SEG;

  #pragma unroll
  for(int i=0;i<(SEG/8)/(32*4);++i){
    int p = wp + (i*32+lane)*4;
    float4 a = *(const float4*)&lx[p];
    float4 c = *(const float4*)&lx[SEG+p];
    float y4[4], yc4[3][4], xc4[3][4], ycv[3], xcv[3];
    pt(a.x,c.x,y4[0],ycv,xcv); for(int o=0;o<3;++o){yc4[o][0]=ycv[o]; xc4[o][0]=xcv[o];}
    pt(a.y,c.y,y4[1],ycv,xcv); for(int o=0;o<3;++o){yc4[o][1]=ycv[o]; xc4[o][1]=xcv[o];}
    pt(a.z,c.z,y4[2],ycv,xcv); for(int o=0;o<3;++o){yc4[o][2]=ycv[o]; xc4[o][2]=xcv[o];}
    pt(a.w,c.w,y4[3],ycv,xcv); for(int o=0;o<3;++o){yc4[o][3]=ycv[o]; xc4[o][3]=xcv[o];}
    *(float4*)(oy+p) = make_float4(y4[0],y4[1],y4[2],y4[3]);
    #pragma unroll
    for(int o=0;o<3;++o){
      *(float4*)(oyc + (size_t)o*N_ + p) = make_float4(yc4[o][0],yc4[o][1],yc4[o][2],yc4[o][3]);
      *(float4*)(oxc + (size_t)o*N_ + p) = make_float4(xc4[o][0],xc4[o][1],xc4[o][2],xc4[o][3]);
    }
  }
}

// =====================================================================
extern "C" void kernel_launch(void* const* d_in, const int* in_sizes, int n_in,
                              void* d_out, int out_size, void* d_ws, size_t ws_size,
                              hipStream_t stream)
{
  (void)in_sizes; (void)n_in; (void)out_size; (void)ws_size;
  const float* x      = (const float*)d_in[0];
  const int*   origin = (const int*)d_in[1];
  const int*   target = (const int*)d_in[2];
  // params leaves in setup_inputs() insertion order:
  // cb0: w0 b0 g0 be0 w1 b1 g1 be1 w2 b2 | sb1: +10 | sb2: +20 | cls: w b
  const float* const* P = reinterpret_cast<const float* const*>(d_in + 3);
  float* ws = (float*)d_ws;

  Consts K;
  K.x = x; K.ws = ws; K.out = (float*)d_out;
  K.c_w0 =P[0];  K.c_b0 =P[1];  K.c_w1 =P[4];  K.c_b1 =P[5];  K.c_w2 =P[8];  K.c_b2 =P[9];
  K.s1_w0=P[10]; K.s1_b0=P[11]; K.s1_w1=P[14]; K.s1_b1=P[15]; K.s1_w2=P[18]; K.s1_b2=P[19];
  K.s2_w0=P[20]; K.s2_b0=P[21]; K.s2_w1=P[24]; K.s2_b1=P[25]; K.s2_w2=P[28]; K.s2_b2=P[29];
  K.cls_w=P[30]; K.cls_b=P[31];

  dim3 g(NBLK), t(TPB);
  k_pass<0><<<g,t,0,stream>>>(K);
  k_fin_bn<<<1,TPB,0,stream>>>(ws, P[2],  P[3],  0);   // cb0 bn0
  k_pass<1><<<g,t,0,stream>>>(K);
  k_fin_bn<<<1,TPB,0,stream>>>(ws, P[6],  P[7],  1);   // cb0 bn1
  k_pass<2><<<g,t,0,stream>>>(K);
  k_fin_gf0<<<1,32,0,stream>>>(ws);
  k_pass<3><<<g,t,0,stream>>>(K);
  k_fin_bn<<<1,TPB,0,stream>>>(ws, P[12], P[13], 2);   // sb1 bn0
  k_pass<4><<<g,t,0,stream>>>(K);
  k_fin_bn<<<1,TPB,0,stream>>>(ws, P[16], P[17], 3);   // sb1 bn1
  k_pass<5><<<g,t,0,stream>>>(K);
  k_fin_bn<<<1,TPB,0,stream>>>(ws, P[22], P[23], 4);   // sb2 bn0
  k_pass<6><<<g,t,0,stream>>>(K);
  k_fin_bn<<<1,TPB,0,stream>>>(ws, P[26], P[27], 5);   // sb2 bn1
  k_rot<<<1,32,0,stream>>>(origin, target, ws);
  k_final<<<g,t,0,stream>>>(K);
}